// RNN_31628139167970
// MI455X (gfx1250) — compile-verified
//
#include <hip/hip_runtime.h>
#include <math.h>

typedef __attribute__((ext_vector_type(2))) float v2f;
typedef __attribute__((ext_vector_type(8))) float v8f;

#define G_CNT 1767
#define T_LEN 6
#define LN    64
#define B_SZ  128

// ---- Branch-free activations (keep EXEC all-ones around the WMMA stream) ----
// gfx1250 has V_TANH_F32; use it if the toolchain exposes the builtin.
#if defined(__has_builtin)
#  if __has_builtin(__builtin_amdgcn_tanhf)
#    define HAVE_HW_TANH 1
#  endif
#endif

__device__ __forceinline__ float tanh_fast(float x) {
#ifdef HAVE_HW_TANH
    return __builtin_amdgcn_tanhf(x);          // single trans op
#else
    // tanh(x) = 1 - 2/(exp(2x)+1); exp->inf => rcp->0 => 1, exp->0 => -1.
    // Entirely branchless: v_exp_f32 + v_rcp_f32 + FMAs.
    float e = __expf(2.0f * x);
    return 1.0f - 2.0f * __builtin_amdgcn_rcpf(e + 1.0f);
#endif
}

__device__ __forceinline__ float sigmoid_fast(float x) {
    // exp(-x)->inf => rcp->0; exp(-x)->0 => 1. Branchless.
    return __builtin_amdgcn_rcpf(1.0f + __expf(-x));
}

// One workgroup (256 threads = 8 wave32) per group g.
// Wave w handles rows rbase=16w .. 16w+15 of the B=128 batch.
// gates(16x256, only i/g/o slices) via V_WMMA_F32_16X16X4_F32, K=6 padded to 2x K=4.
__global__ __launch_bounds__(256) void RNN_31628139167970_kernel(
    const float* __restrict__ x,     // (G,1,B,T)
    const float* __restrict__ Wih,   // (G,256,T)
    const float* __restrict__ bih,   // (G,256)
    const float* __restrict__ bhh,   // (G,256)
    const float* __restrict__ Wreg,  // (G,64)
    const float* __restrict__ breg,  // (G,)
    float* __restrict__ out)         // (G,B) flattened
{
    const int g     = blockIdx.x;
    const int tid   = threadIdx.x;
    const int wave  = tid >> 5;
    const int lane  = tid & 31;
    const int m     = lane & 15;   // column within 16-wide tile (B/C/D N index)
    const int half  = lane >> 4;   // 0: K=0,1  1: K=2,3 (A/B operands)
    const int rbase = wave << 4;

    // ---- A operand: x rows rbase..rbase+15, K=0..5 (pad K=6,7 with 0) ----
    // A layout: lanes 0-15 -> M=lane, {K0,K1}; lanes 16-31 -> M=lane-16, {K2,K3}
    const float2* xr2 =
        reinterpret_cast<const float2*>(x + ((size_t)g * B_SZ + (rbase + m)) * T_LEN);
    float2 xa = xr2[half];                        // half0: (k0,k1)  half1: (k2,k3)
    float2 xz = xr2[2];                           // (k4,k5)
    v2f A0; A0.x = xa.x;                A0.y = xa.y;
    v2f A1; A1.x = half ? 0.0f : xz.x;  A1.y = half ? 0.0f : xz.y;

    const float* wih_g  = Wih  + (size_t)g * 256 * T_LEN;
    const float* bih_g  = bih  + (size_t)g * 256;
    const float* bhh_g  = bhh  + (size_t)g * 256;
    const float* wreg_g = Wreg + (size_t)g * LN;

    // One gate tile: D(16x16) = A(16x6) x Wih[cols col..col+15]^T + bias
    auto gate_gemm = [&](int col) -> v8f {
        // B layout mirrors A: lanes 0-15 -> N=lane, {K0,K1}; lanes 16-31 -> {K2,K3}
        const float2* w2 = reinterpret_cast<const float2*>(wih_g + (size_t)col * T_LEN);
        float2 wb0 = w2[half];
        float2 wb1 = w2[2];
        v2f B0; B0.x = wb0.x;               B0.y = wb0.y;
        v2f B1; B1.x = half ? 0.0f : wb1.x; B1.y = half ? 0.0f : wb1.y;
        // Fold bias into C: C/D element (lane, r) is (M = r + 8*half, N = m),
        // bias depends only on N -> broadcast across the 8 row-VGPRs.
        float bias = bih_g[col] + bhh_g[col];
        v8f c;
        #pragma unroll
        for (int r = 0; r < 8; ++r) c[r] = bias;
        c = __builtin_amdgcn_wmma_f32_16x16x4_f32(false, A0, false, B0,
                                                  (short)0, c, false, false);
        c = __builtin_amdgcn_wmma_f32_16x16x4_f32(false, A1, false, B1,
                                                  (short)0, c, false, false);
        return c;
    };

    float partial[8];
    #pragma unroll
    for (int r = 0; r < 8; ++r) partial[r] = 0.0f;

    // 4 hidden tiles of 16; skip the dead 'f' gate (cols 64..127) entirely.
    #pragma unroll
    for (int jt = 0; jt < 4; ++jt) {
        const int jcol = jt * 16 + m;            // hidden unit index for this lane
        v8f gi = gate_gemm(jcol);                // i-gate cols [0..63]
        v8f gg = gate_gemm(128 + jcol);          // g-gate cols [128..191]
        v8f go = gate_gemm(192 + jcol);          // o-gate cols [192..255]
        const float wr = wreg_g[jcol];
        #pragma unroll
        for (int r = 0; r < 8; ++r) {
            float cc = sigmoid_fast(gi[r]) * tanh_fast(gg[r]);
            float hh = sigmoid_fast(go[r]) * tanh_fast(cc);
            partial[r] += hh * wr;               // row b = rbase + 8*half + r, col jcol
        }
    }

    // Reduce the 16 columns: butterfly over masks 8,4,2,1 stays within each
    // 16-lane half (bit 4 never flips), matching the C/D row split.
    #pragma unroll
    for (int r = 0; r < 8; ++r) {
        float v = partial[r];
        v += __shfl_xor(v, 8, 32);
        v += __shfl_xor(v, 4, 32);
        v += __shfl_xor(v, 2, 32);
        v += __shfl_xor(v, 1, 32);
        partial[r] = v;
    }

    if (m == 0) {
        const float br = breg[g];
        float* op = out + (size_t)g * B_SZ + rbase + half * 8;  // 8 contiguous rows
        #pragma unroll
        for (int r = 0; r < 8; ++r) op[r] = partial[r] + br;
    }
}

extern "C" void kernel_launch(void* const* d_in, const int* in_sizes, int n_in,
                              void* d_out, int out_size, void* d_ws, size_t ws_size,
                              hipStream_t stream) {
    (void)in_sizes; (void)n_in; (void)out_size; (void)d_ws; (void)ws_size;
    const float* x    = (const float*)d_in[0];
    const float* Wih  = (const float*)d_in[1];
    // d_in[2] = W_hh : provably unused by the reference -> never read (saves ~116 MB)
    const float* bih  = (const float*)d_in[3];
    const float* bhh  = (const float*)d_in[4];
    const float* Wreg = (const float*)d_in[5];
    const float* breg = (const float*)d_in[6];
    float* out = (float*)d_out;

    RNN_31628139167970_kernel<<<G_CNT, 256, 0, stream>>>(x, Wih, bih, bhh, Wreg, breg, out);
}